// GNNEncoder_42494406426958
// MI455X (gfx1250) — compile-verified
//
#include <hip/hip_runtime.h>
#include <hip/hip_bf16.h>

typedef __attribute__((ext_vector_type(2))) float v2f;
typedef __attribute__((ext_vector_type(8))) float v8f;

#define D_H 64
#define SLOPE 0.2f
#define GN_EPS 1e-5f

// ---------------------------------------------------------------------------
// f32 atomic add (hardware global_atomic_add_f32, no CAS loop)
// ---------------------------------------------------------------------------
__device__ __forceinline__ void atomic_add_f32(float* p, float v) {
    unsafeAtomicAdd(p, v);
}

// ---------------------------------------------------------------------------
// Degree: deg[dst[e]] += 1
// ---------------------------------------------------------------------------
__global__ __launch_bounds__(256) void degree_kernel(const int* __restrict__ dst,
                                                     float* __restrict__ deg, int e_cnt) {
    int gid = blockIdx.x * blockDim.x + threadIdx.x;
    int stride = gridDim.x * blockDim.x;
    for (int e = gid; e < e_cnt; e += stride)
        atomic_add_f32(&deg[dst[e]], 1.0f);
}

// dinv[i] = rsqrt(deg[i] + 1)   (in place)
__global__ __launch_bounds__(256) void dinv_kernel(float* __restrict__ deg, int n) {
    int i = blockIdx.x * blockDim.x + threadIdx.x;
    if (i < n) deg[i] = rsqrtf(deg[i] + 1.0f);
}

// ---------------------------------------------------------------------------
// GEMM: C[n x 64] = A[n x K] * B[K x 64] (+ optional bias), FP32 WMMA 16x16x4
// 256 threads = 8 waves; block covers 32 rows x 64 cols; 1 tile per wave.
// ---------------------------------------------------------------------------
template <int K>
__global__ __launch_bounds__(256) void gemm_wmma_kernel(const float* __restrict__ A,
                                                        const float* __restrict__ B,
                                                        const float* __restrict__ bias,
                                                        float* __restrict__ C, int n_rows) {
    constexpr int BK = K + 1;   // padded A row stride (breaks bank conflicts)
    constexpr int BN = 64 + 1;  // padded B row stride
    __shared__ float Alds[32 * BK];
    __shared__ float Blds[K * BN];

    const int t = threadIdx.x;
    const int row_base = blockIdx.x * 32;

    // Stage B (K x 64) into LDS, coalesced.
    for (int idx = t; idx < K * 64; idx += 256) {
        int r = idx >> 6, c = idx & 63;
        Blds[r * BN + c] = B[idx];
    }
    // Stage A tile (32 x K) into LDS, coalesced, zero-pad tail rows.
    for (int idx = t; idx < 32 * K; idx += 256) {
        int r = idx / K, c = idx % K;  // K is a power of two -> shifts
        int row = row_base + r;
        Alds[r * BK + c] = (row < n_rows) ? A[(size_t)row * K + c] : 0.0f;
    }
    __syncthreads();

    const int wv   = t >> 5;        // wave id 0..7
    const int lane = t & 31;
    const int rt   = wv >> 2;       // row tile 0..1
    const int ct   = wv & 3;        // col tile 0..3
    const int l15  = lane & 15;
    const int koff = (lane >> 4) << 1;  // lanes 0-15 -> K{0,1}; lanes 16-31 -> K{2,3}

    v8f acc = {};
    const float* Arow = &Alds[(rt * 16 + l15) * BK];
    const float* Bcol = &Blds[ct * 16 + l15];

#pragma unroll
    for (int k = 0; k < K; k += 4) {
        v2f a, b;
        a.x = Arow[k + koff];
        a.y = Arow[k + koff + 1];
        b.x = Bcol[(k + koff) * BN];
        b.y = Bcol[(k + koff + 1) * BN];
        acc = __builtin_amdgcn_wmma_f32_16x16x4_f32(
            /*neg_a=*/false, a, /*neg_b=*/false, b,
            /*c_mod=*/(short)0, acc, /*reuse_a=*/false, /*reuse_b=*/false);
    }

    // Store: VGPR r -> M = r (lanes 0-15) / r+8 (lanes 16-31), N = lane&15.
    const int col  = ct * 16 + l15;
    const int rsub = (lane >> 4) << 3;
    const float bv = bias ? bias[col] : 0.0f;
#pragma unroll
    for (int r = 0; r < 8; ++r) {
        int row = row_base + rt * 16 + r + rsub;
        if (row < n_rows) C[(size_t)row * D_H + col] = acc[r] + bv;
    }
}

// ---------------------------------------------------------------------------
// Edge scatter: agg[dst] += xw[src] * (dinv[src]*dinv[dst]); wave per edge,
// each lane handles 2 features (float2 gather, 2 f32 atomics).
// Edge id is forced into an SGPR (readfirstlane) so the per-edge index and
// coefficient loads become scalar (s_load) instead of 32 redundant vector
// loads; the next edge's gather row is prefetched (global_prefetch_b8) to
// hide the random-access latency behind the current edge's atomics.
// ---------------------------------------------------------------------------
__global__ __launch_bounds__(256) void edge_scatter_kernel(const int* __restrict__ src,
                                                           const int* __restrict__ dst,
                                                           const float* __restrict__ dinv,
                                                           const float* __restrict__ xw,
                                                           float* __restrict__ agg, int e_cnt) {
    const int lane   = threadIdx.x & 31;
    const int wave   = __builtin_amdgcn_readfirstlane(
                           (int)((blockIdx.x * blockDim.x + threadIdx.x) >> 5));
    const int nwaves = (gridDim.x * blockDim.x) >> 5;
    for (int e = wave; e < e_cnt; e += nwaves) {
        int s = src[e], d = dst[e];           // wave-uniform -> scalar loads
        if (e + nwaves < e_cnt) {
            // Prefetch next iteration's gather row (cacheline) into near cache.
            __builtin_prefetch(xw + (size_t)src[e + nwaves] * D_H, 0, 1);
        }
        float c = dinv[s] * dinv[d];          // wave-uniform
        const float2 v = *(const float2*)(xw + (size_t)s * D_H + lane * 2);
        float* p = agg + (size_t)d * D_H + lane * 2;
        atomic_add_f32(p,     v.x * c);
        atomic_add_f32(p + 1, v.y * c);
    }
}

// ---------------------------------------------------------------------------
// Combine conv output in place (agg = agg + xw*dinv^2 + b) and accumulate
// per-feature sum / sum-of-squares for GraphNorm. 64 rows per block.
// ---------------------------------------------------------------------------
__global__ __launch_bounds__(256) void combine_reduce_kernel(float* __restrict__ io_agg,
                                                             const float* __restrict__ xw,
                                                             const float* __restrict__ dinv,
                                                             const float* __restrict__ bias,
                                                             float* __restrict__ sum,
                                                             float* __restrict__ sumsq, int n) {
    __shared__ float s_s[256];
    __shared__ float s_q[256];
    const int t  = threadIdx.x;
    const int d  = t & 63;
    const int rg = t >> 6;  // 0..3
    const int base = blockIdx.x * 64;
    const float bv = bias[d];
    float as = 0.0f, aq = 0.0f;
#pragma unroll 4
    for (int j = 0; j < 16; ++j) {
        int i = base + rg + j * 4;
        if (i < n) {
            float di = dinv[i];
            size_t idx = (size_t)i * D_H + d;
            float v = io_agg[idx] + xw[idx] * di * di + bv;
            io_agg[idx] = v;
            as += v;
            aq += v * v;
        }
    }
    s_s[t] = as;
    s_q[t] = aq;
    __syncthreads();
    if (t < 64) {
        float ts = s_s[t] + s_s[t + 64] + s_s[t + 128] + s_s[t + 192];
        float tq = s_q[t] + s_q[t + 64] + s_q[t + 128] + s_q[t + 192];
        atomic_add_f32(sum + d, ts);
        atomic_add_f32(sumsq + d, tq);
    }
}

// ---------------------------------------------------------------------------
// GraphNorm finalize: per-feature scale/shift.
//  mean = S/n ; var = E[x^2] - (2a - a^2) mean^2
//  out = scale*x + shift,  scale = gamma*rsqrt(var+eps), shift = beta - scale*a*mean
// ---------------------------------------------------------------------------
__global__ __launch_bounds__(64) void finalize_norm_kernel(const float* __restrict__ sum,
                                                           const float* __restrict__ sumsq,
                                                           const float* __restrict__ alpha,
                                                           const float* __restrict__ gamma,
                                                           const float* __restrict__ beta,
                                                           float* __restrict__ AB, int n) {
    int d = threadIdx.x;
    float invN = 1.0f / (float)n;
    float m   = sum[d] * invN;
    float msq = sumsq[d] * invN;
    float a   = alpha[d];
    float var = msq - (2.0f * a - a * a) * m * m;
    float s   = gamma[d] * rsqrtf(var + GN_EPS);
    AB[d]      = s;
    AB[64 + d] = beta[d] - s * a * m;
}

// Elementwise affine (+ optional LeakyReLU), float4-vectorized:
//  out = AB0[d]*in + AB1[d]
template <bool LRELU>
__global__ __launch_bounds__(256) void apply_norm_kernel(const float* __restrict__ in,
                                                         const float* __restrict__ AB,
                                                         float* __restrict__ out, int count4) {
    int gid = blockIdx.x * blockDim.x + threadIdx.x;
    int stride = gridDim.x * blockDim.x;
    const float4* in4 = (const float4*)in;
    float4* out4 = (float4*)out;
    for (int i = gid; i < count4; i += stride) {
        int d = (i << 2) & 63;  // feature index of the first component
        float4 v = in4[i];
        float4 r;
        r.x = AB[d + 0] * v.x + AB[64 + d + 0];
        r.y = AB[d + 1] * v.y + AB[64 + d + 1];
        r.z = AB[d + 2] * v.z + AB[64 + d + 2];
        r.w = AB[d + 3] * v.w + AB[64 + d + 3];
        if (LRELU) {
            r.x = r.x > 0.0f ? r.x : SLOPE * r.x;
            r.y = r.y > 0.0f ? r.y : SLOPE * r.y;
            r.z = r.z > 0.0f ? r.z : SLOPE * r.z;
            r.w = r.w > 0.0f ? r.w : SLOPE * r.w;
        }
        out4[i] = r;
    }
}

// ---------------------------------------------------------------------------
// Launch
// ---------------------------------------------------------------------------
extern "C" void kernel_launch(void* const* d_in, const int* in_sizes, int n_in,
                              void* d_out, int out_size, void* d_ws, size_t ws_size,
                              hipStream_t stream) {
    const float* x      = (const float*)d_in[0];
    const int*   eidx   = (const int*)d_in[1];
    const float* W_in   = (const float*)d_in[2];
    const float* b_in   = (const float*)d_in[3];
    const float* W1     = (const float*)d_in[4];
    const float* b1     = (const float*)d_in[5];
    const float* g1a    = (const float*)d_in[6];
    const float* g1g    = (const float*)d_in[7];
    const float* g1b    = (const float*)d_in[8];
    const float* W2     = (const float*)d_in[9];
    const float* b2     = (const float*)d_in[10];
    const float* g2a    = (const float*)d_in[11];
    const float* g2g    = (const float*)d_in[12];
    const float* g2b    = (const float*)d_in[13];
    float* out = (float*)d_out;

    const int N = in_sizes[0] / 128;  // D_IN = 128
    const int E = in_sizes[1] / 2;
    const size_t N64 = (size_t)N * D_H;

    // Workspace carve-up (floats): h | xw | agg | dinv | sums(128) | AB(128)
    float* ws   = (float*)d_ws;
    float* h    = ws;             // N x 64  (h0, then h1 after GN1+LReLU)
    float* xw   = h + N64;        // N x 64
    float* agg  = xw + N64;       // N x 64  (scatter accum -> conv out in place)
    float* dinv = agg + N64;      // N       (deg -> dinv in place)
    float* sums = dinv + N;       // 64 sum + 64 sumsq
    float* AB   = sums + 128;     // 64 scale + 64 shift

    const int* src = eidx;
    const int* dst = eidx + E;

    const int gemm_blocks = (N + 31) / 32;
    const int comb_blocks = (N + 63) / 64;
    const int elem_blocks = 2048;
    const int edge_blocks = 4096;
    const int count4      = (int)(N64 / 4);

    // --- degrees (shared by both convs) ---
    hipMemsetAsync(dinv, 0, (size_t)N * sizeof(float), stream);
    degree_kernel<<<2048, 256, 0, stream>>>(dst, dinv, E);
    dinv_kernel<<<(N + 255) / 256, 256, 0, stream>>>(dinv, N);

    // --- input projection: h = x @ W_in + b_in ---
    gemm_wmma_kernel<128><<<gemm_blocks, 256, 0, stream>>>(x, W_in, b_in, h, N);

    // ================= Conv 1 =================
    gemm_wmma_kernel<64><<<gemm_blocks, 256, 0, stream>>>(h, W1, nullptr, xw, N);
    hipMemsetAsync(agg, 0, N64 * sizeof(float), stream);
    hipMemsetAsync(sums, 0, 128 * sizeof(float), stream);
    edge_scatter_kernel<<<edge_blocks, 256, 0, stream>>>(src, dst, dinv, xw, agg, E);
    combine_reduce_kernel<<<comb_blocks, 256, 0, stream>>>(agg, xw, dinv, b1, sums, sums + 64, N);
    finalize_norm_kernel<<<1, 64, 0, stream>>>(sums, sums + 64, g1a, g1g, g1b, AB, N);
    apply_norm_kernel<true><<<elem_blocks, 256, 0, stream>>>(agg, AB, h, count4);

    // ================= Conv 2 =================
    gemm_wmma_kernel<64><<<gemm_blocks, 256, 0, stream>>>(h, W2, nullptr, xw, N);
    hipMemsetAsync(agg, 0, N64 * sizeof(float), stream);
    hipMemsetAsync(sums, 0, 128 * sizeof(float), stream);
    edge_scatter_kernel<<<edge_blocks, 256, 0, stream>>>(src, dst, dinv, xw, agg, E);
    combine_reduce_kernel<<<comb_blocks, 256, 0, stream>>>(agg, xw, dinv, b2, sums, sums + 64, N);
    finalize_norm_kernel<<<1, 64, 0, stream>>>(sums, sums + 64, g2a, g2g, g2b, AB, N);
    apply_norm_kernel<false><<<elem_blocks, 256, 0, stream>>>(agg, AB, out, count4);
}